// ProbableConv2d_69741678952811
// MI455X (gfx1250) — compile-verified
//
#include <hip/hip_runtime.h>
#include <math.h>

// Problem constants (match reference)
#define BB   8
#define CIN  64
#define COUT 64
#define NP   9        // number of candidate paths
#define HH   128
#define WW   128
#define HW   (HH * WW)          // 16384
#define NPIX (BB * HW)          // 131072
#define EPSN 1e-5f

typedef __attribute__((ext_vector_type(2))) float v2f;
typedef __attribute__((ext_vector_type(8))) float v8f;

// ---------------------------------------------------------------------------
// CDNA5 async global->LDS copy (ASYNCcnt path), per CDNA5 ISA.
// VDST = VGPR holding LDS byte address; VADDR = 64-bit global address.
// Low 32 bits of a flat pointer into __shared__ are the LDS byte offset.
// ---------------------------------------------------------------------------
__device__ __forceinline__ void async_copy_f32_to_lds(const float* gsrc,
                                                      const float* ldst) {
    unsigned lds_off = (unsigned)(uintptr_t)ldst;
    unsigned long long ga = (unsigned long long)(uintptr_t)gsrc;
    asm volatile("global_load_async_to_lds_b32 %0, %1, off"
                 :
                 : "v"(lds_off), "v"(ga)
                 : "memory");
}

__device__ __forceinline__ void wait_asynccnt0() {
    asm volatile("s_wait_asynccnt 0x0" ::: "memory");
}

// ---------------------------------------------------------------------------
// Kernel 1: observer MLP (1x1 conv -> tanh -> 1x1 conv), per-pixel argmax,
// compaction into per-path pixel lists.
// ---------------------------------------------------------------------------
__global__ __launch_bounds__(256) void observer_argmax_kernel(
    const float* __restrict__ x,      // (B, CIN, H, W)
    const float* __restrict__ w1,     // (32, CIN)
    const float* __restrict__ b1,     // (32,)
    const float* __restrict__ w2,     // (NP, 32)
    const float* __restrict__ b2,     // (NP,)
    int* __restrict__ counts,         // (NP,)  zeroed before launch
    int* __restrict__ lists)          // (NP, NPIX)
{
    int pid = blockIdx.x * 256 + threadIdx.x;
    if (pid >= NPIX) return;

    int b   = pid >> 14;          // / HW
    int off = pid & (HW - 1);
    const float* xb = x + (((size_t)b * CIN) << 14) + off;

    float h[32];
#pragma unroll
    for (int j = 0; j < 32; ++j) h[j] = b1[j];

    for (int ci = 0; ci < CIN; ++ci) {
        float v = xb[(size_t)ci << 14];
#pragma unroll
        for (int j = 0; j < 32; ++j)
            h[j] = fmaf(w1[j * CIN + ci], v, h[j]);
    }
#pragma unroll
    for (int j = 0; j < 32; ++j) h[j] = tanhf(h[j]);

    float best = -INFINITY;
    int bi = 0;
#pragma unroll
    for (int p = 0; p < NP; ++p) {
        float s = b2[p];
#pragma unroll
        for (int j = 0; j < 32; ++j)
            s = fmaf(w2[p * 32 + j], h[j], s);
        if (s > best) { best = s; bi = p; }   // strict > -> first max (jnp.argmax)
    }

    int slot = atomicAdd(&counts[bi], 1);
    lists[bi * NPIX + slot] = pid;
}

// ---------------------------------------------------------------------------
// Kernel 2: selected-path 3x3 conv as gathered implicit GEMM using fp32 WMMA.
// Block = 128 threads (4 waves). Block owns 64 gathered pixels of one path.
// Per wave: 16 pixels (M) x 64 couts (4 N-tiles), K = CIN*9 in 9 async-LDS-
// staged slabs of K=64 (one 3x3 tap each); v_wmma_f32_16x16x4_f32 chains.
// A slab: [pixel][ci], stride 68 -> lane-pair b64 reads conflict-free.
// B slab: K-pair-major float2 [k/2][n], row stride 160 floats (== 32 mod 64
// banks) -> 32-lane b64 reads conflict-free.
// ---------------------------------------------------------------------------
#define A_ST   68      // floats per A row
#define B2_ROW 160     // floats per B k-pair row (64 float2 used + pad)
#define TILES_PER_PATH (NPIX / 64)   // 2048

__global__ __launch_bounds__(128) void path_conv_wmma_kernel(
    const float* __restrict__ x,      // (B, CIN, H, W)
    const float* __restrict__ kw,     // (NP, COUT, CIN, 3, 3)
    const float* __restrict__ kb,     // (NP, COUT)
    const int*  __restrict__ counts,
    const int*  __restrict__ lists,
    const float* __restrict__ zp,     // 64B zeroed scratch (OOB async source)
    float* __restrict__ out)          // (B, COUT, H, W)
{
    int p    = blockIdx.x / TILES_PER_PATH;
    int tile = blockIdx.x % TILES_PER_PATH;
    int cnt  = counts[p];
    int base = tile * 64;
    if (base >= cnt) return;

    __shared__ int pix_s[64];
    __shared__ __align__(16) float a_s[64 * A_ST];     // [pixel][ci]
    __shared__ __align__(16) float b_s[32 * B2_ROW];   // [ci/2][co] float2

    int t = threadIdx.x;
    if (t < 64) {
        int gi = base + t;
        pix_s[t] = (gi < cnt) ? lists[p * NPIX + gi] : -1;
    }
    __syncthreads();

    int lane = t & 31;
    int wv   = t >> 5;                 // wave id 0..3
    int mn   = lane & 15;              // M (A/store) and N (B) index
    int koff = (lane >> 4) << 1;       // lanes 16..31 hold K+2/K+3

    // Accumulators: 4 N-tiles of 16x16 fp32 C/D, initialized with bias.
    v8f acc[4];
#pragma unroll
    for (int nt = 0; nt < 4; ++nt) {
        float bv = kb[p * COUT + nt * 16 + mn];
#pragma unroll
        for (int i = 0; i < 8; ++i) acc[nt][i] = bv;
    }

    // Decode this thread's staging pixel once (2 threads per pixel).
    int st_pix = t >> 1;
    int ci_lo  = (t & 1) * 32;
    int spid   = pix_s[st_pix];
    int sb = 0, sy = 0, sx = 0;
    if (spid >= 0) {
        sb = spid >> 14;
        int o = spid & (HW - 1);
        sy = o >> 7;
        sx = o & (WW - 1);
    }

    for (int r = 0; r < 9; ++r) {      // 3x3 taps
        int ky = r / 3 - 1;
        int kx = r % 3 - 1;

        // --- async-stage weight slab: element (ci, co) of tap r ---
        {
            int co = t >> 1;
            const float* kwp =
                kw + ((size_t)((p * COUT + co) * CIN + ci_lo)) * 9 + r;
#pragma unroll
            for (int i = 0; i < 32; ++i) {
                int ci = ci_lo + i;
                async_copy_f32_to_lds(
                    kwp + (size_t)i * 9,
                    &b_s[(ci >> 1) * B2_ROW + co * 2 + (ci & 1)]);
            }
        }
        // --- async-stage im2col slab: a_s[pixel][ci] for tap r ---
        {
            int yy = sy + ky, xx = sx + kx;
            bool valid = (spid >= 0) & (yy >= 0) & (yy < HH) &
                         (xx >= 0) & (xx < WW);
            const float* xp =
                x + (((size_t)sb * CIN) << 14) + (yy << 7) + xx;
#pragma unroll
            for (int i = 0; i < 32; ++i) {
                int ci = ci_lo + i;
                const float* gsrc = valid ? (xp + ((size_t)ci << 14)) : zp;
                async_copy_f32_to_lds(gsrc, &a_s[st_pix * A_ST + ci]);
            }
        }
        wait_asynccnt0();
        __syncthreads();

        // --- WMMA over this K=64 slab ---
        int arow = (wv * 16 + mn) * A_ST;
#pragma unroll 4
        for (int kk = 0; kk < 64; kk += 4) {
            v2f af = *(const v2f*)&a_s[arow + kk + koff];       // ds_load_b64
            const float* brow = &b_s[((kk + koff) >> 1) * B2_ROW];
#pragma unroll
            for (int nt = 0; nt < 4; ++nt) {
                v2f bf = *(const v2f*)&brow[(nt * 16 + mn) * 2]; // ds_load_b64
                acc[nt] = __builtin_amdgcn_wmma_f32_16x16x4_f32(
                    /*neg_a=*/false, af, /*neg_b=*/false, bf,
                    /*c_mod=*/(short)0, acc[nt],
                    /*reuse_a=*/false, /*reuse_b=*/false);
            }
        }
        __syncthreads();
    }

    // --- store relu'd results (scatter by pixel id) ---
#pragma unroll
    for (int i = 0; i < 8; ++i) {
        int m    = i + ((lane >> 4) << 3);   // M of this accumulator row
        int slot = wv * 16 + m;
        int pid  = pix_s[slot];
        if (pid >= 0) {
            int b   = pid >> 14;
            int off = pid & (HW - 1);
#pragma unroll
            for (int nt = 0; nt < 4; ++nt) {
                int co = nt * 16 + mn;
                out[(((size_t)(b * COUT + co)) << 14) + off] =
                    fmaxf(acc[nt][i], 0.0f);
            }
        }
    }
}

// ---------------------------------------------------------------------------
// Kernel 3a: per-(b,c) mean and rsqrt(var+eps) over the relu'd conv output.
// ---------------------------------------------------------------------------
__global__ __launch_bounds__(256) void stats_kernel(
    const float* __restrict__ pre, float* __restrict__ mu, float* __restrict__ rv)
{
    int bc = blockIdx.x;               // 0..B*COUT-1
    const float* ptr = pre + ((size_t)bc << 14);
    float s = 0.0f, s2 = 0.0f;
    for (int i = threadIdx.x; i < HW; i += 256) {
        float v = ptr[i];
        s += v; s2 += v * v;
    }
    __shared__ float ss[256], ss2[256];
    ss[threadIdx.x] = s; ss2[threadIdx.x] = s2;
    __syncthreads();
    for (int st = 128; st > 0; st >>= 1) {
        if (threadIdx.x < st) {
            ss[threadIdx.x]  += ss[threadIdx.x + st];
            ss2[threadIdx.x] += ss2[threadIdx.x + st];
        }
        __syncthreads();
    }
    if (threadIdx.x == 0) {
        float mean = ss[0] * (1.0f / HW);
        float var  = ss2[0] * (1.0f / HW) - mean * mean;
        mu[bc] = mean;
        rv[bc] = rsqrtf(var + EPSN);
    }
}

// ---------------------------------------------------------------------------
// Kernel 3b: in-place normalize + affine.
// ---------------------------------------------------------------------------
__global__ __launch_bounds__(256) void norm_kernel(
    float* __restrict__ out, const float* __restrict__ mu,
    const float* __restrict__ rv, const float* __restrict__ gamma,
    const float* __restrict__ beta)
{
    size_t i = (size_t)blockIdx.x * 256 + threadIdx.x;
    int bc = (int)(i >> 14);
    int c  = bc & (COUT - 1);
    float v = out[i];
    out[i] = (v - mu[bc]) * rv[bc] * gamma[c] + beta[c];
}

// ---------------------------------------------------------------------------
extern "C" void kernel_launch(void* const* d_in, const int* in_sizes, int n_in,
                              void* d_out, int out_size, void* d_ws, size_t ws_size,
                              hipStream_t stream) {
    const float* x   = (const float*)d_in[0];
    const float* kw  = (const float*)d_in[1];
    const float* kb  = (const float*)d_in[2];
    const float* w1  = (const float*)d_in[3];
    const float* b1  = (const float*)d_in[4];
    const float* w2  = (const float*)d_in[5];
    const float* b2  = (const float*)d_in[6];
    const float* gamma = (const float*)d_in[7];
    const float* beta  = (const float*)d_in[8];
    float* out = (float*)d_out;

    // Workspace layout:
    //   [0,   36)  : path counts (9 ints)
    //   [128, 256) : zeroed scratch (async-OOB source)
    //   [256, ...) : per-path pixel lists, then mu/rv stats
    char* ws = (char*)d_ws;
    int*         counts = (int*)ws;
    const float* zp     = (const float*)(ws + 128);
    int*         lists  = (int*)(ws + 256);
    float*       mu     = (float*)(ws + 256 + (size_t)NP * NPIX * sizeof(int));
    float*       rv     = mu + BB * COUT;

    hipMemsetAsync(ws, 0, 256, stream);   // clears counts AND the zero slot

    observer_argmax_kernel<<<NPIX / 256, 256, 0, stream>>>(
        x, w1, b1, w2, b2, counts, lists);

    path_conv_wmma_kernel<<<NP * TILES_PER_PATH, 128, 0, stream>>>(
        x, kw, kb, counts, lists, zp, out);

    stats_kernel<<<BB * COUT, 256, 0, stream>>>(out, mu, rv);

    norm_kernel<<<(NPIX * COUT) / 256, 256, 0, stream>>>(out, mu, rv, gamma, beta);
}